// SineGenerator_20779051778640
// MI455X (gfx1250) — compile-verified
//
#include <hip/hip_runtime.h>
#include <hip/hip_bf16.h>
#include <math.h>
#include <stdint.h>

#define DIMH 9
#define SRATE 24000.0f
#define SINE_AMPC 0.1f
#define NOISE_STDC 0.003f
#define UNVOICED_AMP (SINE_AMPC / 3.0f)
#define TWO_PI_F 6.283185307179586f

// gfx1250 async global->LDS path (probe-confirmed: arg0 = addrspace(1) int*,
// arg1 = addrspace(3) int*).
#if defined(__HIP_DEVICE_COMPILE__) && defined(__has_builtin)
#if __has_builtin(__builtin_amdgcn_global_load_async_to_lds_b32) && \
    __has_builtin(__builtin_amdgcn_s_wait_asynccnt)
#define HAVE_ASYNC_LDS 1
#endif
#endif

#if HAVE_ASYNC_LDS
typedef __attribute__((address_space(1))) int as1_int;
typedef __attribute__((address_space(3))) int as3_int;
#endif

// ---------------------------------------------------------------------------
// Stage 1: frame-level prefix sums. One wave32 per (b, harmonic) sequence.
// E[b,n,d]   = frac-reduced exclusive prefix of rad over frames  (phase mod 1)
// RAD[b,n,d] = rad value per frame (rand_phase folded into frame 0)
// 72 sequences x 1000 frames: negligible next to stage 2; keep it simple.
// ---------------------------------------------------------------------------
__global__ void sine_stage1(const float* __restrict__ f0,
                            const float* __restrict__ rand_phase,
                            float* __restrict__ E,
                            float* __restrict__ RAD,
                            int T) {
  const int b    = blockIdx.x / DIMH;
  const int d    = blockIdx.x % DIMH;
  const int lane = threadIdx.x;  // blockDim.x == 32 (one wave)

  const float mult_over_sr = (float)(d + 1) * (1.0f / SRATE);
  const float rp = (d == 0) ? 0.0f : rand_phase[b * DIMH + d];

  float carry = 0.0f;
  for (int base = 0; base < T; base += 32) {
    const int n = base + lane;
    float f = (n < T) ? f0[b * T + n] : 0.0f;
    float r = f * mult_over_sr;
    r -= floorf(r);              // f0 >= 0, matches (x % 1.0)
    if (n == 0) r += rp;         // reference adds rand_phase without re-mod

    // wave32 inclusive scan (Hillis-Steele via shfl_up)
    float v = r;
#pragma unroll
    for (int off = 1; off < 32; off <<= 1) {
      float u = __shfl_up(v, off, 32);
      if (lane >= off) v += u;
    }

    if (n < T) {
      const size_t idx = ((size_t)b * T + n) * DIMH + d;
      E[idx]   = carry + (v - r);  // exclusive prefix
      RAD[idx] = r;
    }
    carry += __shfl(v, 31, 32);
    carry -= floorf(carry);        // keep carry bounded: only phase mod 1 matters
  }
}

// ---------------------------------------------------------------------------
// Stage 2: one block per (b, frame n); blockDim.x == U (240) output samples.
// phase(t) = E[b,n,d] + (k+1)*rad[b,n,d]   (integer reset shifts cancel in sin)
// Streaming traffic (~276 MB) uses non-temporal TH policy so the one-touch
// noise/output streams don't evict the reused E/RAD/f0 working set from L2.
// ---------------------------------------------------------------------------
__global__ void sine_stage2(const float* __restrict__ f0,
                            const float* __restrict__ noise_randn,
                            const float* __restrict__ E,
                            const float* __restrict__ RAD,
                            float* __restrict__ out,
                            int T, int U, int Tout, long long N1) {
  const int b = blockIdx.y;
  const int n = blockIdx.x;
  const int k = threadIdx.x;
  const int frame = b * T + n;

  __shared__ float sER[19];  // [0..8]=E, [9..17]=rad, [18]=f0(frame)

#if HAVE_ASYNC_LDS
  if (k < 19) {
    const float* g = (k < 9)    ? (E   + (size_t)frame * DIMH + k)
                   : (k < 18)   ? (RAD + (size_t)frame * DIMH + (k - 9))
                                : (f0  + frame);
    __builtin_amdgcn_global_load_async_to_lds_b32(
        (as1_int*)(void*)g,          // global src (addrspace(1) int*)
        (as3_int*)&sER[k],           // LDS dst    (addrspace(3) int*)
        /*offset=*/0, /*cpol=*/0);
  }
  __builtin_amdgcn_s_wait_asynccnt(0);
#else
  if (k < 19) {
    sER[k] = (k < 9)  ? E[(size_t)frame * DIMH + k]
           : (k < 18) ? RAD[(size_t)frame * DIMH + (k - 9)]
                      : f0[frame];
  }
#endif
  __syncthreads();

  const float uv   = (sER[18] > 0.0f) ? 1.0f : 0.0f;
  const float namp = (uv > 0.0f) ? NOISE_STDC : UNVOICED_AMP;
  const float kp1  = (float)(k + 1);

  const long long t    = (long long)n * U + k;
  const size_t    base = ((size_t)b * Tout + t) * DIMH;

  const float* __restrict__ nr  = noise_randn + base;
  float* __restrict__ o_sine    = out + base;
  float* __restrict__ o_uv      = out + (size_t)N1 + base;
  float* __restrict__ o_noise   = out + (size_t)(2 * N1) + base;

#pragma unroll
  for (int d = 0; d < DIMH; ++d) {
    float ph = sER[d] + kp1 * sER[9 + d];
    ph -= floorf(ph);                              // phase mod 1 (f32-safe)
    const float s  = __sinf(TWO_PI_F * ph) * SINE_AMPC;  // v_sin_f32 path
    const float nv = namp * __builtin_nontemporal_load(nr + d);  // th:NT read-once
    __builtin_nontemporal_store(s * uv + nv, o_sine + d);        // th:NT streams
    __builtin_nontemporal_store(uv,          o_uv + d);
    __builtin_nontemporal_store(nv,          o_noise + d);
  }
}

// ---------------------------------------------------------------------------
extern "C" void kernel_launch(void* const* d_in, const int* in_sizes, int n_in,
                              void* d_out, int out_size, void* d_ws, size_t ws_size,
                              hipStream_t stream) {
  const float* f0 = (const float*)d_in[0];           // (B, T)
  const float* rp = (const float*)d_in[1];           // (B, DIM)
  const float* nz = (const float*)d_in[2];           // (B, T*U, DIM)

  const int B = in_sizes[1] / DIMH;
  const int T = in_sizes[0] / B;
  const int U = (int)((long long)in_sizes[2] / ((long long)in_sizes[0] * DIMH));
  const int Tout = T * U;
  const long long N1 = (long long)B * Tout * DIMH;

  float* E   = (float*)d_ws;                         // B*T*DIM floats
  float* RAD = E + (size_t)B * T * DIMH;             // B*T*DIM floats

  sine_stage1<<<B * DIMH, 32, 0, stream>>>(f0, rp, E, RAD, T);

  dim3 grid(T, B);
  sine_stage2<<<grid, U, 0, stream>>>(f0, nz, E, RAD, (float*)d_out,
                                      T, U, Tout, N1);

  (void)n_in; (void)out_size; (void)ws_size;
}